// GCNEncoder_14688788152881
// MI455X (gfx1250) — compile-verified
//
#include <hip/hip_runtime.h>
#include <hip/hip_bf16.h>

// ---------------------------------------------------------------------------
// GCN 2-layer forward for MI455X (gfx1250, wave32).
//  - Dense transform via V_WMMA_F32_16X16X4_F32 (fp32 exact, GEMM is tiny).
//  - Edge scatter via global_atomic_add_f32; h table (25.6MB) lives in the
//    192MB L2, so gather/scatter is L2-bound (~0.7GB/layer of L2 traffic).
//  - Self-loop term folded into bias+ReLU epilogue (no atomics for it).
// ---------------------------------------------------------------------------

typedef __attribute__((ext_vector_type(2))) float v2f;
typedef __attribute__((ext_vector_type(8))) float v8f;

#define C_DIM 128   // both layers are 128 channels

__global__ void fill_zero_f32(float* __restrict__ p, int n) {
  int i = blockIdx.x * blockDim.x + threadIdx.x;
  if (i < n) p[i] = 0.0f;
}

// deg[d] += 1 per incoming edge (self-loop added later in dinv_kernel)
__global__ void deg_kernel(const int* __restrict__ dst, float* __restrict__ deg, int E) {
  int i = blockIdx.x * blockDim.x + threadIdx.x;
  if (i < E) atomicAdd(&deg[dst[i]], 1.0f);
}

// deg -> 1/sqrt(deg+1), in place (deg+1 >= 1, never zero)
__global__ void dinv_kernel(float* __restrict__ deg, int n) {
  int i = blockIdx.x * blockDim.x + threadIdx.x;
  if (i < n) deg[i] = rsqrtf(deg[i] + 1.0f);
}

// H[N,128] = X[N,128] @ W[128,128] — one wave per 16x16 output tile,
// K swept 4 at a time with V_WMMA_F32_16X16X4_F32.
// A lane layout (16x4 f32): lane L holds row M=L%16; lanes 0-15 carry K=k,k+1,
// lanes 16-31 carry K=k+2,k+3. B (4x16) mirrors with N=L%16.
// D layout: VGPR r holds (M = r + 8*(L/16), N = L%16).
__global__ void gemm_wmma_f32(const float* __restrict__ X, const float* __restrict__ W,
                              float* __restrict__ H, int nRows) {
  const int row  = blockIdx.x * 16;          // node-tile base
  const int col  = blockIdx.y * 16;          // channel-tile base
  const int lane = threadIdx.x;              // 0..31 (wave32)
  const int half = lane >> 4;                // 0 or 1
  const int idx  = lane & 15;

  int r0 = row + idx;
  if (r0 > nRows - 1) r0 = nRows - 1;        // clamp (keeps EXEC all-1 for WMMA)
  const float* xrow = X + (size_t)r0 * C_DIM;

  v8f c = {};
#pragma unroll
  for (int k = 0; k < C_DIM; k += 4) {
    const int kk = k + 2 * half;             // even -> 8B-aligned float2 load
    v2f a, b;
    const float2 av = *(const float2*)(xrow + kk);
    a.x = av.x; a.y = av.y;
    b.x = W[(size_t)kk * C_DIM + col + idx];
    b.y = W[(size_t)(kk + 1) * C_DIM + col + idx];
    c = __builtin_amdgcn_wmma_f32_16x16x4_f32(
        /*neg_a=*/false, a, /*neg_b=*/false, b,
        /*c_mod=*/(short)0, c, /*reuse_a=*/false, /*reuse_b=*/false);
  }

#pragma unroll
  for (int r = 0; r < 8; ++r) {
    int rs = row + r + 8 * half;
    if (rs < nRows) H[(size_t)rs * C_DIM + col + idx] = c[r];
  }
}

// One wave per edge: out[dst] += dinv[src]*dinv[dst] * h[src], 128 channels
// as 32 lanes x float4, via global_atomic_add_f32 (L2 atomic units).
__global__ void scatter_edges(const float* __restrict__ H, const int* __restrict__ src,
                              const int* __restrict__ dst, const float* __restrict__ dinv,
                              float* __restrict__ out, int E) {
  const int wave = (blockIdx.x * blockDim.x + threadIdx.x) >> 5;
  const int lane = threadIdx.x & 31;
  if (wave >= E) return;
  const int s = src[wave];
  const int d = dst[wave];
  const float nrm = dinv[s] * dinv[d];
  const float4 v = ((const float4*)(H + (size_t)s * C_DIM))[lane];
  float* od = out + (size_t)d * C_DIM + lane * 4;
  atomicAdd(od + 0, v.x * nrm);
  atomicAdd(od + 1, v.y * nrm);
  atomicAdd(od + 2, v.z * nrm);
  atomicAdd(od + 3, v.w * nrm);
}

// Epilogue: out = relu(out + dinv[i]^2 * h[i] + bias[c])  (self-loop + bias)
__global__ void fuse_epilogue(float* __restrict__ out, const float* __restrict__ H,
                              const float* __restrict__ dinv, const float* __restrict__ bias,
                              int total) {
  int i = blockIdx.x * blockDim.x + threadIdx.x;
  if (i < total) {
    const int node = i >> 7;      // /128
    const int c    = i & (C_DIM - 1);
    const float di = dinv[node];
    const float v  = out[i] + di * di * H[i] + bias[c];
    out[i] = v > 0.0f ? v : 0.0f;
  }
}

extern "C" void kernel_launch(void* const* d_in, const int* in_sizes, int n_in,
                              void* d_out, int out_size, void* d_ws, size_t ws_size,
                              hipStream_t stream) {
  const float* x  = (const float*)d_in[0];
  const int*   ei = (const int*)  d_in[1];   // [2,E]: row0=src, row1=dst
  const float* W1 = (const float*)d_in[2];
  const float* b1 = (const float*)d_in[3];
  const float* W2 = (const float*)d_in[4];
  const float* b2 = (const float*)d_in[5];
  float* out = (float*)d_out;

  const int N = in_sizes[0] / C_DIM;
  const int E = in_sizes[1] / 2;
  const int* src = ei;
  const int* dst = ei + E;

  // workspace: dinv (N floats, 256B-aligned pad) + one N*128 feature buffer
  const size_t off = (((size_t)N * sizeof(float)) + 255) & ~(size_t)255;
  float* dinv = (float*)d_ws;
  float* h    = (float*)((char*)d_ws + off);

  const int TB = 256;
  const int total = N * C_DIM;
  const int nBlk  = (N + TB - 1) / TB;
  const int eBlk  = (E + TB - 1) / TB;
  const int tBlk  = (total + TB - 1) / TB;
  const int sBlk  = (E + 7) / 8;            // 8 waves of 32 per block, 1 edge/wave
  dim3 ggrid((N + 15) / 16, C_DIM / 16);

  // --- normalization: dinv = 1/sqrt(deg_in + 1) ---
  fill_zero_f32<<<nBlk, TB, 0, stream>>>(dinv, N);
  deg_kernel  <<<eBlk, TB, 0, stream>>>(dst, dinv, E);
  dinv_kernel <<<nBlk, TB, 0, stream>>>(dinv, N);

  // --- layer 1: h = x@W1 ; out = relu(scatter(h) + self + b1) ---
  gemm_wmma_f32<<<ggrid, 32, 0, stream>>>(x, W1, h, N);
  fill_zero_f32<<<tBlk, TB, 0, stream>>>(out, total);
  scatter_edges<<<sBlk, TB, 0, stream>>>(h, src, dst, dinv, out, E);
  fuse_epilogue<<<tBlk, TB, 0, stream>>>(out, h, dinv, b1, total);

  // --- layer 2: h = out@W2 ; out = relu(scatter(h) + self + b2) ---
  gemm_wmma_f32<<<ggrid, 32, 0, stream>>>(out, W2, h, N);
  fill_zero_f32<<<tBlk, TB, 0, stream>>>(out, total);
  scatter_edges<<<sBlk, TB, 0, stream>>>(h, src, dst, dinv, out, E);
  fuse_epilogue<<<tBlk, TB, 0, stream>>>(out, h, dinv, b2, total);
}